// DynamicRouter_68410239090949
// MI455X (gfx1250) — compile-verified
//
#include <hip/hip_runtime.h>
#include <math.h>

typedef __attribute__((ext_vector_type(2))) float v2f;
typedef __attribute__((ext_vector_type(8))) float v8f;

#define D_DIM 4096
#define E_DIM 64
#define ROWS_PER_BLOCK 64    // 4 waves * 16 rows
#define THREADS 128          // 4 wave32s
#define N_TOKENS (4 * 4096)  // B*S
#define EPS_F 1e-5f

__global__ __launch_bounds__(THREADS) void router_wmma_kernel(
    const float* __restrict__ x,      // [N_TOKENS, D]
    const float* __restrict__ gw,     // [E, D] (row-major; logits = x @ gw^T)
    const float* __restrict__ tptr,   // [1] temperature
    float* __restrict__ rout,         // [N_TOKENS, E]
    int* __restrict__ idx_out)        // [N_TOKENS, 2]
{
    // +1 padding column -> bank-conflict-free row reads in the top-k phase
    __shared__ float lds[ROWS_PER_BLOCK * 65];
    __shared__ float w0s[ROWS_PER_BLOCK], w1s[ROWS_PER_BLOCK];
    __shared__ int   i0s[ROWS_PER_BLOCK], i1s[ROWS_PER_BLOCK];

    const int tid  = threadIdx.x;
    const int wv   = tid >> 5;        // wave id 0..3 (wave32)
    const int lane = tid & 31;
    const int m16  = lane & 15;       // row-within-tile / col-within-tile
    const int h    = lane >> 4;       // lane half selects K pair (A/B) and M half (C)
    const int rowBase = blockIdx.x * ROWS_PER_BLOCK + wv * 16;

    // A fragment (16x4 f32): lane l, vgpr r -> A[m16][k + r + 2h]
    const float* ap = x  + (size_t)(rowBase + m16) * D_DIM + 2 * h;
    // B fragment (4x16 f32): lane l, vgpr r -> B[k + r + 2h][n=m16] = gw[col][k + r + 2h]
    const float* bp = gw + (size_t)m16 * D_DIM + 2 * h;

    v8f acc0 = {}, acc1 = {}, acc2 = {}, acc3 = {};

    for (int k0 = 0; k0 < D_DIM; k0 += 32) {
        // pull the streamed A lines toward the WGP ahead of the b64 consumers
        if (k0 + 1024 < D_DIM) {
            __builtin_prefetch(ap + k0 + 1024, 0, 0);   // global_prefetch_b8
        }
        #pragma unroll
        for (int kk = 0; kk < 32; kk += 4) {
            const int k = k0 + kk;
            // X is streamed once -> non-temporal (don't thrash L2 where gate_w lives)
            v2f a  = __builtin_nontemporal_load((const v2f*)(ap + k));
            // gate_w (1 MB) is re-read by every wave -> regular temporal, stays in L2
            v2f b0 = *(const v2f*)(bp + k);
            v2f b1 = *(const v2f*)(bp + k + 16 * D_DIM);
            v2f b2 = *(const v2f*)(bp + k + 32 * D_DIM);
            v2f b3 = *(const v2f*)(bp + k + 48 * D_DIM);
            acc0 = __builtin_amdgcn_wmma_f32_16x16x4_f32(false, a, false, b0, (short)0, acc0, false, false);
            acc1 = __builtin_amdgcn_wmma_f32_16x16x4_f32(false, a, false, b1, (short)0, acc1, false, false);
            acc2 = __builtin_amdgcn_wmma_f32_16x16x4_f32(false, a, false, b2, (short)0, acc2, false, false);
            acc3 = __builtin_amdgcn_wmma_f32_16x16x4_f32(false, a, false, b3, (short)0, acc3, false, false);
        }
    }

    // C/D layout: vgpr r, lane l -> m = r + 8*(l>>4), n = l&15
    #pragma unroll
    for (int r = 0; r < 8; ++r) {
        const int row = wv * 16 + r + 8 * h;
        lds[row * 65 + m16 +  0] = acc0[r];
        lds[row * 65 + m16 + 16] = acc1[r];
        lds[row * 65 + m16 + 32] = acc2[r];
        lds[row * 65 + m16 + 48] = acc3[r];
    }
    __syncthreads();

    // ---- top-2 + softmax per token (one token per thread) ----
    const float scale = 1.0f / (fabsf(tptr[0]) + EPS_F);
    if (tid < ROWS_PER_BLOCK) {
        const float* rowp = &lds[tid * 65];
        float v0 = rowp[0]; int i0 = 0;
        #pragma unroll
        for (int e = 1; e < E_DIM; ++e) {
            float v = rowp[e];
            if (v > v0) { v0 = v; i0 = e; }      // strict > => lowest index on tie
        }
        float v1 = -INFINITY; int i1 = 0;
        #pragma unroll
        for (int e = 0; e < E_DIM; ++e) {
            float v = rowp[e];
            if (e != i0 && v > v1) { v1 = v; i1 = e; }
        }
        // softmax over the two scaled logits; v0 >= v1 so this is max-subtracted
        const float p1  = __expf((v1 - v0) * scale);
        const float inv = 1.0f / (1.0f + p1);
        w0s[tid] = inv;
        w1s[tid] = p1 * inv;
        i0s[tid] = i0;
        i1s[tid] = i1;

        const int gtok = blockIdx.x * ROWS_PER_BLOCK + tid;
        idx_out[gtok * 2 + 0] = i0;
        idx_out[gtok * 2 + 1] = i1;
    }
    __syncthreads();

    // ---- coalesced dense routing-matrix write (written once -> non-temporal) ----
    const size_t outBase = (size_t)blockIdx.x * (ROWS_PER_BLOCK * E_DIM);
    #pragma unroll
    for (int j = 0; j < (ROWS_PER_BLOCK * E_DIM) / THREADS; ++j) {
        const int flat = j * THREADS + tid;
        const int row  = flat >> 6;
        const int e    = flat & 63;
        float val = 0.0f;
        if (e == i0s[row]) val = w0s[row];
        else if (e == i1s[row]) val = w1s[row];
        __builtin_nontemporal_store(val, &rout[outBase + flat]);
    }
}

extern "C" void kernel_launch(void* const* d_in, const int* in_sizes, int n_in,
                              void* d_out, int out_size, void* d_ws, size_t ws_size,
                              hipStream_t stream) {
    const float* x    = (const float*)d_in[0];  // [4,4096,4096] f32
    const float* gw   = (const float*)d_in[1];  // [64,4096] f32
    const float* temp = (const float*)d_in[2];  // scalar f32

    // d_out = routing_matrix [4,4096,64] f32, then top_idx [4,4096,2] i32
    float* routp  = (float*)d_out;
    int*   idxOut = (int*)((float*)d_out + (size_t)N_TOKENS * E_DIM);

    dim3 grid(N_TOKENS / ROWS_PER_BLOCK);  // 256 blocks
    dim3 block(THREADS);                   // 4 wave32s
    router_wmma_kernel<<<grid, block, 0, stream>>>(x, gw, temp, routp, idxOut);
}